// GCN_18580028523179
// MI455X (gfx1250) — compile-verified
//
#include <hip/hip_runtime.h>

typedef __attribute__((ext_vector_type(2))) float v2f;
typedef __attribute__((ext_vector_type(8))) float v8f;

#define D     64
#define TROWS 32
#define LDP   68   // LDS row pitch (floats): 64 + 4 pad -> conflict-free column reads

// ---------------- degree / normalization ----------------
__global__ void gcn_deg_init(float* deg, int n) {
  int i = blockIdx.x * blockDim.x + threadIdx.x;
  if (i < n) deg[i] = 1.0f;                      // self-loop weight 1
}

__global__ void gcn_deg_edges(float* deg, const int* __restrict__ dst,
                              const float* __restrict__ ew, int e) {
  int i = blockIdx.x * blockDim.x + threadIdx.x;
  if (i < e) atomicAdd(&deg[dst[i]], ew[i]);
}

__global__ void gcn_dinv(float* deg, int n) {
  int i = blockIdx.x * blockDim.x + threadIdx.x;
  if (i < n) {
    float d = deg[i];
    deg[i] = (d > 0.0f) ? rsqrtf(d) : 0.0f;      // deg buffer becomes dinv
  }
}

// ---------------- feature staging ----------------
__global__ void gcn_copy_x(const float* __restrict__ x, float* __restrict__ Hcat, int n) {
  int i = blockIdx.x * blockDim.x + threadIdx.x;
  if (i < n * D) {
    int node = i >> 6, j = i & 63;
    Hcat[(size_t)node * 256 + j] = x[i];
  }
}

// out_slice[i,:] = b + dinv[i]^2 * h[i,:]   (self-loop contribution + bias)
__global__ void gcn_agg_self(const float* __restrict__ Htmp, const float* __restrict__ dinv,
                             const float* __restrict__ bias, float* __restrict__ Hcat,
                             int off, int n) {
  int i = blockIdx.x * blockDim.x + threadIdx.x;
  if (i < n * D) {
    int node = i >> 6, j = i & 63;
    float di = dinv[node];
    Hcat[(size_t)node * 256 + off + j] = bias[j] + di * di * Htmp[i];
  }
}

// 16 lanes per edge, float4 gather of h[src], 4x global_atomic_add_f32 into dst row
__global__ void gcn_scatter(const float* __restrict__ Htmp, const float* __restrict__ dinv,
                            const float* __restrict__ ew, const int* __restrict__ src,
                            const int* __restrict__ dst, float* __restrict__ Hcat,
                            int off, int e) {
  int t = blockIdx.x * blockDim.x + threadIdx.x;
  int eid = t >> 4;
  int l   = t & 15;
  if (eid < e) {
    int s = src[eid], d = dst[eid];
    float nrm = dinv[s] * ew[eid] * dinv[d];
    const float4 h = *(const float4*)(Htmp + (size_t)s * D + l * 4);
    float* o = Hcat + (size_t)d * 256 + off + l * 4;
    atomicAdd(o + 0, h.x * nrm);
    atomicAdd(o + 1, h.y * nrm);
    atomicAdd(o + 2, h.z * nrm);
    atomicAdd(o + 3, h.w * nrm);
  }
}

__global__ void gcn_relu(float* Hcat, int off, int n) {
  int i = blockIdx.x * blockDim.x + threadIdx.x;
  if (i < n * D) {
    int node = i >> 6, j = i & 63;
    size_t idx = (size_t)node * 256 + off + j;
    float v = Hcat[idx];
    Hcat[idx] = v > 0.0f ? v : 0.0f;
  }
}

// ---------------- fp32 WMMA GEMM: Out[n,64] = X[n,K](pitch ldx) @ W[K,64] (+bias) ----
// 256 threads = 8 wave32s; each wave owns one 16x16 tile of a 32x64 block tile.
// Chained V_WMMA_F32_16X16X4_F32 over K in 64-wide LDS-staged chunks.
__global__ void gcn_gemm_wmma(const float* __restrict__ X, int ldx,
                              const float* __restrict__ W,
                              const float* __restrict__ bias,
                              float* __restrict__ Out, int K, int n) {
  __shared__ float Xs[TROWS * LDP];
  __shared__ float Ws[D * LDP];
  const int tid    = threadIdx.x;
  const int wave   = tid >> 5;
  const int lane   = tid & 31;
  const int laneLo = lane & 15;
  const int laneHi = lane >> 4;
  const int row0   = blockIdx.x * TROWS;
  const int wm     = (wave >> 2) << 4;   // 0 / 16 within block tile
  const int wn     = (wave & 3) << 4;    // 0 / 16 / 32 / 48

  v8f acc = {};
  for (int kc = 0; kc < K; kc += D) {
    // stage X tile (32 x 64) coalesced
    for (int i = tid; i < TROWS * D; i += 256) {
      int r = i >> 6, c = i & 63;
      int rr = row0 + r; if (rr >= n) rr = n - 1;
      Xs[r * LDP + c] = X[(size_t)rr * ldx + kc + c];
    }
    // stage W tile (64 x 64) coalesced
    for (int i = tid; i < D * D; i += 256) {
      int r = i >> 6, c = i & 63;
      Ws[r * LDP + c] = W[(size_t)(kc + r) * D + c];
    }
    __syncthreads();
#pragma unroll
    for (int k0 = 0; k0 < D; k0 += 4) {
      // A frag: lane -> (M = laneLo, K = k0 + 2*laneHi + v), contiguous -> one b64 LDS load
      v2f a = *(const v2f*)(&Xs[(wm + laneLo) * LDP + k0 + 2 * laneHi]);
      // B frag: lane -> (K = k0 + 2*laneHi + v, N = laneLo)
      v2f b;
      b[0] = Ws[(k0 + 2 * laneHi + 0) * LDP + wn + laneLo];
      b[1] = Ws[(k0 + 2 * laneHi + 1) * LDP + wn + laneLo];
      acc = __builtin_amdgcn_wmma_f32_16x16x4_f32(false, a, false, b,
                                                  (short)0, acc, false, false);
    }
    __syncthreads();
  }
  const int gcol = wn + laneLo;
  const float bv = bias ? bias[gcol] : 0.0f;
#pragma unroll
  for (int v = 0; v < 8; ++v) {   // D layout: M = v + 8*laneHi
    int grow = row0 + wm + v + 8 * laneHi;
    if (grow < n) Out[(size_t)grow * D + gcol] = acc[v] + bv;
  }
}

// ---------------- launch ----------------
extern "C" void kernel_launch(void* const* d_in, const int* in_sizes, int n_in,
                              void* d_out, int out_size, void* d_ws, size_t ws_size,
                              hipStream_t stream) {
  const float* x   = (const float*)d_in[0];
  const int*   ei  = (const int*)d_in[1];     // [2,E] row-major: src then dst
  const float* ew  = (const float*)d_in[2];
  const float* Wl[3] = {(const float*)d_in[3], (const float*)d_in[5], (const float*)d_in[7]};
  const float* bl[3] = {(const float*)d_in[4], (const float*)d_in[6], (const float*)d_in[8]};
  const float* Wf  = (const float*)d_in[9];
  const float* bf  = (const float*)d_in[10];

  const int N = in_sizes[0] / D;
  const int E = in_sizes[2];
  const int* src = ei;
  const int* dst = ei + E;

  // workspace: dinv[N] | Hcat[N,256] | Htmp[N,64]  (~128.4 MB total)
  float* deg  = (float*)d_ws;
  float* Hcat = deg + (((size_t)N + 255) & ~(size_t)255);
  float* Htmp = Hcat + (size_t)N * 256;

  const int T = 256;
  gcn_deg_init <<<(N + T - 1) / T, T, 0, stream>>>(deg, N);
  gcn_deg_edges<<<(E + T - 1) / T, T, 0, stream>>>(deg, dst, ew, E);
  gcn_dinv     <<<(N + T - 1) / T, T, 0, stream>>>(deg, N);
  gcn_copy_x   <<<((size_t)N * D + T - 1) / T, T, 0, stream>>>(x, Hcat, N);

  const int gemm_blocks = (N + TROWS - 1) / TROWS;
  for (int l = 0; l < 3; ++l) {
    const int off = (l + 1) * D;
    gcn_gemm_wmma<<<gemm_blocks, T, 0, stream>>>(Hcat + l * D, 256, Wl[l], nullptr,
                                                 Htmp, D, N);
    gcn_agg_self <<<((size_t)N * D + T - 1) / T, T, 0, stream>>>(Htmp, deg, bl[l],
                                                                 Hcat, off, N);
    gcn_scatter  <<<((size_t)E * 16 + T - 1) / T, T, 0, stream>>>(Htmp, deg, ew, src, dst,
                                                                  Hcat, off, E);
    gcn_relu     <<<((size_t)N * D + T - 1) / T, T, 0, stream>>>(Hcat, off, N);
  }
  // final: d_out = Hcat(N x 256) @ Wf(256 x 64) + bf
  gcn_gemm_wmma<<<gemm_blocks, T, 0, stream>>>(Hcat, 256, Wf, bf, (float*)d_out, 4 * D, N);
}